// FactorizedVectorQuantize_34076270527017
// MI455X (gfx1250) — compile-verified
//
#include <hip/hip_runtime.h>

#define B_    8
#define DIM_  1024
#define T_    4096
#define CB_   4096
#define CD_   8
#define BT_   (B_ * T_)

typedef __attribute__((ext_vector_type(2))) float v2f;
typedef __attribute__((ext_vector_type(8))) float v8f;

// ---------------- K1: W_in = weight_norm(v_in, g_in)  [CD_ x DIM_] ----------------
__global__ void k_win(const float* __restrict__ v_in, const float* __restrict__ g_in,
                      float* __restrict__ Win) {
    __shared__ float red[256];
    const int o = blockIdx.x, tid = threadIdx.x;
    float ss = 0.f;
    for (int d = tid; d < DIM_; d += 256) { float v = v_in[o * DIM_ + d]; ss += v * v; }
    red[tid] = ss; __syncthreads();
    for (int s = 128; s > 0; s >>= 1) { if (tid < s) red[tid] += red[tid + s]; __syncthreads(); }
    const float scale = g_in[o] / sqrtf(red[0]);
    for (int d = tid; d < DIM_; d += 256) Win[o * DIM_ + d] = v_in[o * DIM_ + d] * scale;
}

// ---------------- K2: c_n = l2_normalize(codebook)  [CB_ x CD_] ----------------
__global__ void k_cn(const float* __restrict__ cb, float* __restrict__ cn) {
    const int c = blockIdx.x * 256 + threadIdx.x;
    const float4* p = (const float4*)(cb + (size_t)c * CD_);
    float4 a = p[0], b = p[1];
    float n2 = a.x*a.x + a.y*a.y + a.z*a.z + a.w*a.w + b.x*b.x + b.y*b.y + b.z*b.z + b.w*b.w;
    float rn = 1.f / fmaxf(sqrtf(n2), 1e-12f);
    a.x*=rn; a.y*=rn; a.z*=rn; a.w*=rn; b.x*=rn; b.y*=rn; b.z*=rn; b.w*=rn;
    float4* q = (float4*)(cn + (size_t)c * CD_);
    q[0] = a; q[1] = b;
}

// ---------------- K3: P[code][o] = sum_c codebook[code][c] * W_out[o][c] ----------------
__global__ void k_projcb(const float* __restrict__ cb, const float* __restrict__ v_out,
                         const float* __restrict__ g_out, float* __restrict__ P) {
    const int o = blockIdx.x * 256 + threadIdx.x;   // grid.x = 4
    const int code = blockIdx.y;                    // grid.y = CB_
    const float4* vp = (const float4*)(v_out + (size_t)o * CD_);
    float4 va = vp[0], vb = vp[1];
    float n2 = va.x*va.x + va.y*va.y + va.z*va.z + va.w*va.w
             + vb.x*vb.x + vb.y*vb.y + vb.z*vb.z + vb.w*vb.w;
    const float s = g_out[o] / sqrtf(n2);
    const float4* cp = (const float4*)(cb + (size_t)code * CD_);   // uniform -> scalar loads
    float4 ca = cp[0], cbv = cp[1];
    float d = va.x*ca.x + va.y*ca.y + va.z*ca.z + va.w*ca.w
            + vb.x*cbv.x + vb.y*cbv.y + vb.z*cbv.z + vb.w*cbv.w;
    P[(size_t)code * DIM_ + o] = s * d;
}

// ---------------- K4: z_e[token][c] = W_in . z(:,t) + b_in  (token-major) ----------------
__global__ void k_ze(const float* __restrict__ z, const float* __restrict__ Win,
                     const float* __restrict__ b_in, float* __restrict__ ze) {
    __shared__ float sW[DIM_ * CD_];                 // [d][o] transposed, 32 KB
    const int b = blockIdx.x;
    const int t = blockIdx.y * 256 + threadIdx.x;
    for (int i = threadIdx.x; i < DIM_ * CD_; i += 256) {
        int o = i / DIM_, d = i % DIM_;
        sW[d * CD_ + o] = Win[i];
    }
    __syncthreads();
    float acc[CD_] = {0.f, 0.f, 0.f, 0.f, 0.f, 0.f, 0.f, 0.f};
    const float* zp = z + (size_t)b * DIM_ * T_ + t;
    for (int d = 0; d < DIM_; ++d) {
        if ((d & 31) == 0 && d + 32 < DIM_)
            __builtin_prefetch(zp + (size_t)(d + 32) * T_, 0, 0);   // global_prefetch_b8
        float zv = zp[(size_t)d * T_];               // coalesced along t
        const float4* w = (const float4*)&sW[d * CD_]; // LDS broadcast
        float4 w0 = w[0], w1 = w[1];
        acc[0] = fmaf(w0.x, zv, acc[0]); acc[1] = fmaf(w0.y, zv, acc[1]);
        acc[2] = fmaf(w0.z, zv, acc[2]); acc[3] = fmaf(w0.w, zv, acc[3]);
        acc[4] = fmaf(w1.x, zv, acc[4]); acc[5] = fmaf(w1.y, zv, acc[5]);
        acc[6] = fmaf(w1.z, zv, acc[6]); acc[7] = fmaf(w1.w, zv, acc[7]);
    }
    const size_t tok = (size_t)b * T_ + t;
    float4* o4 = (float4*)(ze + tok * CD_);
    o4[0] = make_float4(acc[0] + b_in[0], acc[1] + b_in[1], acc[2] + b_in[2], acc[3] + b_in[3]);
    o4[1] = make_float4(acc[4] + b_in[4], acc[5] + b_in[5], acc[6] + b_in[6], acc[7] + b_in[7]);
}

// ---------------- K5: WMMA code search with async-LDS double-buffered codebook --------
// GEMM M=BT_, N=CB_, K=8 via two V_WMMA_F32_16X16X4_F32 per 16x16 tile, running argmax.
// Codebook staged through LDS in 1024-code chunks with GLOBAL_LOAD_ASYNC_TO_LDS_B128,
// double-buffered and synchronized with s_wait_asynccnt + workgroup barrier.
#define SEARCH_WAVES 8
#define CHUNK_CODES  1024
#define ROW_F        12      // 8 data floats + 4 pad: 48B rows, 16B aligned, bank-conflict-free

__global__ __launch_bounds__(256) void k_search(const float* __restrict__ ze,
                                                const float* __restrict__ cn,
                                                int* __restrict__ idxOut) {
    __shared__ __align__(16) float sB[2][CHUNK_CODES * ROW_F];   // 2 x 48 KB
    const int tid  = threadIdx.x;
    const int lane = tid & 31;
    const int wave = tid >> 5;
    const int tb   = (blockIdx.x * SEARCH_WAVES + wave) * 16;
    const int m    = lane & 15;        // M-row (A) / N-col (B/D) within tile
    const int hi   = lane >> 4;        // K-half selector

    const float2* af = (const float2*)(ze + (size_t)(tb + m) * CD_);
    float2 al = af[hi], ah = af[2 + hi];
    v2f A0 = {al.x, al.y};             // K = {0,1} | {2,3}
    v2f A1 = {ah.x, ah.y};             // K = {4,5} | {6,7}

    float bestV[8]; int bestI[8];
#pragma unroll
    for (int r = 0; r < 8; ++r) { bestV[r] = -3.402823e38f; bestI[r] = 0; }

    auto issue_chunk = [&](int chunk, int buf) {
        // 1024 rows * 32B = 2048 x B128 transfers; 8 per thread
#pragma unroll
        for (int j = 0; j < 8; ++j) {
            const int e    = j * 256 + tid;
            const int row  = e >> 1;
            const int part = e & 1;
            const float* gp = cn + (size_t)chunk * (CHUNK_CODES * CD_) + row * CD_ + part * 4;
            const unsigned lp =
                (unsigned)(uintptr_t)(const void*)&sB[buf][row * ROW_F + part * 4];
            asm volatile("global_load_async_to_lds_b128 %0, %1, off"
                         :: "v"(lp), "v"(gp) : "memory");
        }
    };

    issue_chunk(0, 0);
    asm volatile("s_wait_asynccnt 0x0" ::: "memory");
    __syncthreads();

    for (int chunk = 0; chunk < CB_ / CHUNK_CODES; ++chunk) {
        const int buf = chunk & 1;
        if (chunk + 1 < CB_ / CHUNK_CODES) issue_chunk(chunk + 1, buf ^ 1);  // overlap DMA
        const float* bbase = &sB[buf][0];
        const int cbase = chunk * CHUNK_CODES;
        for (int nt = 0; nt < CHUNK_CODES / 16; ++nt) {
            const int cl = nt * 16 + m;
            const float2* bf = (const float2*)(bbase + cl * ROW_F);
            float2 bl = bf[hi], bh = bf[2 + hi];                 // ds_load_b64, no conflicts
            v2f B0 = {bl.x, bl.y};
            v2f B1 = {bh.x, bh.y};
            v8f acc = {};
            acc = __builtin_amdgcn_wmma_f32_16x16x4_f32(false, A0, false, B0, (short)0, acc, false, false);
            acc = __builtin_amdgcn_wmma_f32_16x16x4_f32(false, A1, false, B1, (short)0, acc, false, false);
#pragma unroll
            for (int r = 0; r < 8; ++r) {        // D: VGPR r -> M = r + 8*hi, N = m
                float v = acc[r];
                if (v > bestV[r]) { bestV[r] = v; bestI[r] = cbase + cl; }  // strict >: lowest idx wins
            }
        }
        asm volatile("s_wait_asynccnt 0x0" ::: "memory");
        __syncthreads();   // next buffer ready AND this buffer free to overwrite
    }

#pragma unroll
    for (int r = 0; r < 8; ++r) {
        float v = bestV[r]; int bi = bestI[r];
#pragma unroll
        for (int off = 1; off <= 8; off <<= 1) {   // reduce within 16-lane halves
            float ov = __shfl_xor(v, off, 32);
            int   oi = __shfl_xor(bi, off, 32);
            if (ov > v || (ov == v && oi < bi)) { v = ov; bi = oi; }
        }
        if (m == 0) idxOut[tb + r + 8 * hi] = bi;
    }
}

// ---------------- K6: per-token loss partials + float indices ----------------
__global__ void k_token(const float* __restrict__ ze, const float* __restrict__ cb,
                        const int* __restrict__ idx, float* __restrict__ idxF,
                        float* __restrict__ partial) {
    __shared__ float red[256];
    const int token = blockIdx.x * 256 + threadIdx.x;
    const int id = idx[token];
    const float4* zp = (const float4*)(ze + (size_t)token * CD_);
    const float4* cp = (const float4*)(cb + (size_t)id * CD_);
    float4 z0 = zp[0], z1 = zp[1], c0 = cp[0], c1 = cp[1];
    float d, s = 0.f;
    d = z0.x - c0.x; s += d*d;  d = z0.y - c0.y; s += d*d;
    d = z0.z - c0.z; s += d*d;  d = z0.w - c0.w; s += d*d;
    d = z1.x - c1.x; s += d*d;  d = z1.y - c1.y; s += d*d;
    d = z1.z - c1.z; s += d*d;  d = z1.w - c1.w; s += d*d;
    idxF[token] = (float)id;
    red[threadIdx.x] = s; __syncthreads();
    for (int st = 128; st > 0; st >>= 1) { if (threadIdx.x < st) red[threadIdx.x] += red[threadIdx.x + st]; __syncthreads(); }
    if (threadIdx.x == 0) partial[blockIdx.x] = red[0];
}

// ---------------- K7: deterministic loss finalize ----------------
__global__ void k_loss(const float* __restrict__ partial, float* __restrict__ loss) {
    const int b = threadIdx.x;
    if (b < B_) {
        float s = 0.f;
        for (int j = 0; j < 16; ++j) s += partial[b * 16 + j];
        loss[b] = 1.25f * s / (float)(CD_ * T_);   // (1 + 0.25) * mean over (CD,T)
    }
}

// ---------------- K8: out[b,o,t] = P[idx[b,t]][o] + b_out[o]  (LDS transpose tile) ----------------
__global__ void k_out(const float* __restrict__ P, const int* __restrict__ idx,
                      const float* __restrict__ b_out, float* __restrict__ out) {
    __shared__ int   sIdx[64];
    __shared__ float tile[64][65];
    const int b  = blockIdx.z;
    const int t0 = blockIdx.x * 64;
    const int o0 = blockIdx.y * 64;
    const int tid = threadIdx.x;
    if (tid < 64) sIdx[tid] = idx[b * T_ + t0 + tid];
    __syncthreads();
    const int tIdx = tid & 63, og = tid >> 6;
    const size_t row = (size_t)sIdx[tIdx] * DIM_;
#pragma unroll
    for (int j = 0; j < 4; ++j) {
        int ol = (og * 4 + j) * 4;
        float4 p = *(const float4*)(P + row + o0 + ol);   // L2-resident gather
        tile[ol + 0][tIdx] = p.x; tile[ol + 1][tIdx] = p.y;
        tile[ol + 2][tIdx] = p.z; tile[ol + 3][tIdx] = p.w;
    }
    __syncthreads();
    const int tl = tid & 63;
#pragma unroll
    for (int j = 0; j < 16; ++j) {
        int ol = (tid >> 6) + j * 4;
        out[((size_t)(b * DIM_ + o0 + ol)) * T_ + t0 + tl] = tile[ol][tl] + b_out[o0 + ol];
    }
}

extern "C" void kernel_launch(void* const* d_in, const int* in_sizes, int n_in,
                              void* d_out, int out_size, void* d_ws, size_t ws_size,
                              hipStream_t stream) {
    const float* z        = (const float*)d_in[0];
    const float* v_in     = (const float*)d_in[1];
    const float* g_in     = (const float*)d_in[2];
    const float* b_in     = (const float*)d_in[3];
    const float* codebook = (const float*)d_in[4];
    const float* v_out    = (const float*)d_in[5];
    const float* g_out    = (const float*)d_in[6];
    const float* b_out    = (const float*)d_in[7];

    // workspace partition (all 256B-aligned)
    char* w = (char*)d_ws;
    float* Win     = (float*)w;  w += (size_t)CD_ * DIM_ * 4;      // 32 KB
    float* cn      = (float*)w;  w += (size_t)CB_ * CD_ * 4;       // 128 KB
    float* ze      = (float*)w;  w += (size_t)BT_ * CD_ * 4;       // 1 MB
    int*   idx     = (int*)w;    w += (size_t)BT_ * 4;             // 128 KB
    float* partial = (float*)w;  w += 256 * 4;                     // 1 KB
    float* P       = (float*)w;                                    // 16 MB

    float* outMain = (float*)d_out;                                // [B, DIM, T]
    float* outIdx  = outMain + (size_t)B_ * DIM_ * T_;             // [B, T] as float
    float* outLoss = outIdx + (size_t)BT_;                         // [B]

    k_win   <<<CD_,                          256, 0, stream>>>(v_in, g_in, Win);
    k_cn    <<<CB_ / 256,                    256, 0, stream>>>(codebook, cn);
    k_projcb<<<dim3(DIM_ / 256, CB_),        256, 0, stream>>>(codebook, v_out, g_out, P);
    k_ze    <<<dim3(B_, T_ / 256),           256, 0, stream>>>(z, Win, b_in, ze);
    k_search<<<BT_ / (16 * SEARCH_WAVES),    256, 0, stream>>>(ze, cn, idx);
    k_token <<<BT_ / 256,                    256, 0, stream>>>(ze, codebook, idx, outIdx, partial);
    k_loss  <<<1,                            32,  0, stream>>>(partial, outLoss);
    k_out   <<<dim3(T_ / 64, DIM_ / 64, B_), 256, 0, stream>>>(P, idx, b_out, outMain);
}